// Model_80985903333896
// MI455X (gfx1250) — compile-verified
//
#include <hip/hip_runtime.h>

#define B_   8
#define M_   4096
#define W_   128
#define C_   32
#define NN_  16
#define J_   4
#define ED_  5
#define R_   (B_ * M_)          // 32768 rows (b,m)
#define K1_  (M_ * C_)          // 131072 fc1 reduction dim
#define EPS_ 1e-5f
#define FC1SPLIT 32
#define FC1CHUNK (K1_ / FC1SPLIT)

typedef float v2f __attribute__((ext_vector_type(2)));
typedef float v8f __attribute__((ext_vector_type(8)));

// ---------------------------------------------------------------------------
// 1. Edge attention: Gaussian mixture weight per edge (both layers), softmax
//    over each row's 16 neighbors, scatter-set duplicate semantics.
// ---------------------------------------------------------------------------
__global__ void edge_attn_kernel(const float* __restrict__ pseudo,
                                 const int* __restrict__ L_idx,
                                 const float* __restrict__ ew,
                                 const float* __restrict__ eb,
                                 const float* __restrict__ mu0,
                                 const float* __restrict__ sg0,
                                 const float* __restrict__ mu1,
                                 const float* __restrict__ sg1,
                                 float* __restrict__ attn0,
                                 float* __restrict__ attn1) {
  __shared__ float sw0[256], sw1[256];
  __shared__ int   snb[256];
  const int tid = threadIdx.x;
  const int e   = blockIdx.x * 256 + tid;

  const float p0 = pseudo[e * 2 + 0];
  const float p1 = pseudo[e * 2 + 1];
  float emb[ED_];
#pragma unroll
  for (int d = 0; d < ED_; ++d)
    emb[d] = ew[d * 2] * p0 + ew[d * 2 + 1] * p1 + eb[d];

  float w0 = 0.f, w1 = 0.f;
#pragma unroll
  for (int j = 0; j < J_; ++j) {
    float q0 = 0.f, q1 = 0.f;
#pragma unroll
    for (int d = 0; d < ED_; ++d) {
      float u0 = emb[d] - mu0[j * ED_ + d]; q0 += u0 * u0 * sg0[j * ED_ + d];
      float u1 = emb[d] - mu1[j * ED_ + d]; q1 += u1 * u1 * sg1[j * ED_ + d];
    }
    w0 += __expf(-0.5f * q0);
    w1 += __expf(-0.5f * q1);
  }

  const int m = e / NN_;
  snb[tid] = L_idx[e] - m * M_;   // recover neighbor index
  sw0[tid] = w0;
  sw1[tid] = w1;
  __syncthreads();

  const int g    = tid & ~(NN_ - 1);
  const int lane = tid & (NN_ - 1);
  float mx0 = -1e30f, mx1 = -1e30f;
#pragma unroll
  for (int i = 0; i < NN_; ++i) { mx0 = fmaxf(mx0, sw0[g + i]); mx1 = fmaxf(mx1, sw1[g + i]); }
  float s0 = 0.f, s1 = 0.f;
#pragma unroll
  for (int i = 0; i < NN_; ++i) { s0 += __expf(sw0[g + i] - mx0); s1 += __expf(sw1[g + i] - mx1); }
  float a0 = __expf(w0 - mx0) / s0;
  float a1 = __expf(w1 - mx1) / s1;

  // .at[idx].set(): last occurrence of a duplicate neighbor wins
  const int my = snb[tid];
  bool keep = true;
  for (int i = lane + 1; i < NN_; ++i)
    if (snb[g + i] == my) keep = false;
  if (!keep) { a0 = 0.f; a1 = 0.f; }

  attn0[e] = a0;
  attn1[e] = a1;
}

// ---------------------------------------------------------------------------
// 2. Sparse graph conv gather: Lx[b,m,:] = sum_nn attn[m,nn] * x[b,nbr,:]
// ---------------------------------------------------------------------------
template <int F>
__global__ void gather_kernel(const float* __restrict__ xin,
                              const float* __restrict__ attn,
                              const int* __restrict__ L_idx,
                              float* __restrict__ Lx) {
  __shared__ float sa[NN_];
  __shared__ int   snb[NN_];
  const int m   = blockIdx.x;
  const int tid = threadIdx.x;
  if (tid < NN_) {
    sa[tid]  = attn[m * NN_ + tid];
    snb[tid] = L_idx[m * NN_ + tid] - m * M_;
  }
  __syncthreads();
  const int f = tid % F;
  for (int b = tid / F; b < B_; b += blockDim.x / F) {
    float acc = 0.f;
#pragma unroll
    for (int nn = 0; nn < NN_; ++nn)
      acc += sa[nn] * xin[((size_t)b * M_ + snb[nn]) * F + f];
    Lx[((size_t)b * M_ + m) * F + f] = acc;
  }
}

// ---------------------------------------------------------------------------
// 3. Dual GEMM via V_WMMA_F32_16X16X4_F32:
//    out[r,c] = sum_k A0[r,k]*W1[c,k] + sum_k A1[r,k]*W2[c,k] + b1[c] + b2[c]
//    Block = 64 threads = 2 waves; each wave owns one 16x16 N-tile.
// ---------------------------------------------------------------------------
template <int K>
__global__ void gemm2_kernel(const float* __restrict__ A0,   // Lx   (R x K)
                             const float* __restrict__ A1,   // x/h  (R x K)
                             const float* __restrict__ W1,   // (C x K)
                             const float* __restrict__ b1,
                             const float* __restrict__ W2,   // (C x K)
                             const float* __restrict__ b2,
                             float* __restrict__ out) {      // (R x C)
  const int row0    = blockIdx.x * 16;
  const int wave    = threadIdx.x >> 5;
  const int lane    = threadIdx.x & 31;
  const int colbase = wave * 16;
  const int m16     = lane & 15;
  const int khalf   = (lane >> 4) * 2;          // lanes 16-31 carry K = k0+2, k0+3
  const int n       = colbase + m16;

  const float* a0row = A0 + (size_t)(row0 + m16) * K;
  const float* a1row = A1 + (size_t)(row0 + m16) * K;
  const float* w1row = W1 + (size_t)n * K;
  const float* w2row = W2 + (size_t)n * K;

  v8f acc = {};
#pragma unroll
  for (int k0 = 0; k0 < K; k0 += 4) {
    const int kk = k0 + khalf;
    v2f a, b;
    a.x = a0row[kk]; a.y = a0row[kk + 1];
    b.x = w1row[kk]; b.y = w1row[kk + 1];
    acc = __builtin_amdgcn_wmma_f32_16x16x4_f32(false, a, false, b,
                                                (short)0, acc, false, false);
    a.x = a1row[kk]; a.y = a1row[kk + 1];
    b.x = w2row[kk]; b.y = w2row[kk + 1];
    acc = __builtin_amdgcn_wmma_f32_16x16x4_f32(false, a, false, b,
                                                (short)0, acc, false, false);
  }

  // D layout: VGPR r -> (M = r [+8 for lanes 16-31], N = lane & 15)
  const int   col  = colbase + (lane & 15);
  const int   rofs = (lane >> 4) * 8;
  const float bias = b1[col] + b2[col];
#pragma unroll
  for (int r = 0; r < 8; ++r) {
    const int row = row0 + rofs + r;
    out[(size_t)row * C_ + col] = acc[r] + bias;
  }
}

// ---------------------------------------------------------------------------
// 4. BatchNorm (training mode): per-channel sum / sumsq, then normalize+ReLU
// ---------------------------------------------------------------------------
__global__ void zero_kernel(float* __restrict__ p, int n) {
  const int i = blockIdx.x * 256 + threadIdx.x;
  if (i < n) p[i] = 0.f;
}

__global__ void bn_stats_kernel(const float* __restrict__ h,
                                float* __restrict__ stats) {  // [0:32)=sum [32:64)=sumsq
  __shared__ float ssum[256], ssq[256];
  const int tid = threadIdx.x;
  const int c   = tid & 31;
  const int rs  = tid >> 5;                 // 0..7
  const size_t rbase = (size_t)blockIdx.x * 64;
  float sum = 0.f, sq = 0.f;
  for (int i = rs; i < 64; i += 8) {
    const float v = h[(rbase + i) * C_ + c];
    sum += v; sq += v * v;
  }
  ssum[tid] = sum; ssq[tid] = sq;
  __syncthreads();
  if (rs == 0) {
#pragma unroll
    for (int i = 1; i < 8; ++i) { sum += ssum[i * 32 + c]; sq += ssq[i * 32 + c]; }
    atomicAdd(&stats[c], sum);
    atomicAdd(&stats[32 + c], sq);
  }
}

__global__ void bn_apply_kernel(float* __restrict__ h,
                                const float* __restrict__ stats,
                                const float* __restrict__ g,
                                const float* __restrict__ bb) {
  const size_t i = (size_t)blockIdx.x * 256 + threadIdx.x;   // over R*C
  const int c = (int)(i & 31);
  const float mean = stats[c] * (1.f / R_);
  const float var  = stats[32 + c] * (1.f / R_) - mean * mean;
  const float sc   = g[c] * rsqrtf(var + EPS_);
  const float v    = (h[i] - mean) * sc + bb[c];
  h[i] = fmaxf(v, 0.f);
}

// ---------------------------------------------------------------------------
// 5. fc1 split-K: bandwidth-bound GEMV over the 134 MB weight matrix.
//    h1 layout (b*M + m)*C + c  ==  b*K1 + k with k = m*C + c  (matches ref flatten)
// ---------------------------------------------------------------------------
__global__ void fc1_kernel(const float* __restrict__ h,
                           const float* __restrict__ w,
                           float* __restrict__ partial) {     // [256][8]
  __shared__ float red[256];
  const int o   = blockIdx.y;
  const int tid = threadIdx.x;
  const int kbase = blockIdx.x * FC1CHUNK;
  float acc[B_];
#pragma unroll
  for (int b = 0; b < B_; ++b) acc[b] = 0.f;
  const float* wrow = w + (size_t)o * K1_;
  for (int k = kbase + tid; k < kbase + FC1CHUNK; k += 256) {
    const float wv = wrow[k];
#pragma unroll
    for (int b = 0; b < B_; ++b)
      acc[b] += wv * h[(size_t)b * K1_ + k];
  }
#pragma unroll
  for (int b = 0; b < B_; ++b) {
    red[tid] = acc[b];
    __syncthreads();
    for (int s = 128; s > 0; s >>= 1) {
      if (tid < s) red[tid] += red[tid + s];
      __syncthreads();
    }
    if (tid == 0) atomicAdd(&partial[o * B_ + b], red[0]);
    __syncthreads();
  }
}

// ---------------------------------------------------------------------------
// 6. fc1 bias + ReLU, then fc2 (256 -> 53) in one block
// ---------------------------------------------------------------------------
__global__ void fc_final_kernel(const float* __restrict__ partial,
                                const float* __restrict__ b1,
                                const float* __restrict__ w2,
                                const float* __restrict__ b2,
                                float* __restrict__ out) {
  __shared__ float a[B_ * 256];
  const int tid = threadIdx.x;                 // 512 threads
  for (int i = tid; i < B_ * 256; i += 512) {
    const int b = i >> 8, o = i & 255;
    a[i] = fmaxf(partial[o * B_ + b] + b1[o], 0.f);
  }
  __syncthreads();
  if (tid < B_ * 53) {
    const int b = tid / 53, j = tid % 53;
    float s = b2[j];
#pragma unroll 4
    for (int o = 0; o < 256; ++o) s += a[b * 256 + o] * w2[j * 256 + o];
    out[b * 53 + j] = s;
  }
}

// ---------------------------------------------------------------------------
extern "C" void kernel_launch(void* const* d_in, const int* in_sizes, int n_in,
                              void* d_out, int out_size, void* d_ws, size_t ws_size,
                              hipStream_t stream) {
  (void)in_sizes; (void)n_in; (void)out_size; (void)ws_size;
  const float* x      = (const float*)d_in[0];
  const float* pseudo = (const float*)d_in[1];
  const int*   L_idx  = (const int*)d_in[2];
  const float* edge_w = (const float*)d_in[3];
  const float* edge_b = (const float*)d_in[4];
  const float* mu0    = (const float*)d_in[5];
  const float* sg0    = (const float*)d_in[6];
  const float* mu1    = (const float*)d_in[7];
  const float* sg1    = (const float*)d_in[8];
  const float* l1w0   = (const float*)d_in[9];
  const float* l1b0   = (const float*)d_in[10];
  const float* l2w0   = (const float*)d_in[11];
  const float* l2b0   = (const float*)d_in[12];
  const float* l1w1   = (const float*)d_in[13];
  const float* l1b1   = (const float*)d_in[14];
  const float* l2w1   = (const float*)d_in[15];
  const float* l2b1   = (const float*)d_in[16];
  const float* bn_g0  = (const float*)d_in[17];
  const float* bn_b0  = (const float*)d_in[18];
  const float* bn_g1  = (const float*)d_in[19];
  const float* bn_b1  = (const float*)d_in[20];
  const float* fc1w   = (const float*)d_in[21];
  const float* fc1b   = (const float*)d_in[22];
  const float* fc2w   = (const float*)d_in[23];
  const float* fc2b   = (const float*)d_in[24];
  float* out = (float*)d_out;

  // workspace carve-out (~29 MB)
  char* ws = (char*)d_ws;
  size_t off = 0;
  auto carve = [&](size_t elems) {
    float* p = (float*)(ws + off);
    off += ((elems * sizeof(float) + 255) / 256) * 256;
    return p;
  };
  float* attn0 = carve((size_t)M_ * NN_);
  float* attn1 = carve((size_t)M_ * NN_);
  float* Lx0   = carve((size_t)R_ * W_);
  float* Lx1   = carve((size_t)R_ * C_);
  float* h0    = carve((size_t)R_ * C_);
  float* h1    = carve((size_t)R_ * C_);
  float* stats = carve(64);
  float* part  = carve(256 * B_);

  edge_attn_kernel<<<(M_ * NN_) / 256, 256, 0, stream>>>(
      pseudo, L_idx, edge_w, edge_b, mu0, sg0, mu1, sg1, attn0, attn1);

  // ---- layer 0 (W=128 -> C=32) ----
  gather_kernel<W_><<<M_, 128, 0, stream>>>(x, attn0, L_idx, Lx0);
  gemm2_kernel<W_><<<R_ / 16, 64, 0, stream>>>(Lx0, x, l1w0, l1b0, l2w0, l2b0, h0);
  zero_kernel<<<1, 256, 0, stream>>>(stats, 64);
  bn_stats_kernel<<<R_ / 64, 256, 0, stream>>>(h0, stats);
  bn_apply_kernel<<<(R_ * C_) / 256, 256, 0, stream>>>(h0, stats, bn_g0, bn_b0);

  // ---- layer 1 (C=32 -> C=32) ----
  gather_kernel<C_><<<M_, 128, 0, stream>>>(h0, attn1, L_idx, Lx1);
  gemm2_kernel<C_><<<R_ / 16, 64, 0, stream>>>(Lx1, h0, l1w1, l1b1, l2w1, l2b1, h1);
  zero_kernel<<<1, 256, 0, stream>>>(stats, 64);
  bn_stats_kernel<<<R_ / 64, 256, 0, stream>>>(h1, stats);
  bn_apply_kernel<<<(R_ * C_) / 256, 256, 0, stream>>>(h1, stats, bn_g1, bn_b1);

  // ---- classifier head ----
  zero_kernel<<<(256 * B_ + 255) / 256, 256, 0, stream>>>(part, 256 * B_);
  fc1_kernel<<<dim3(FC1SPLIT, 256), 256, 0, stream>>>(h1, fc1w, part);
  fc_final_kernel<<<1, 512, 0, stream>>>(part, fc1b, fc2w, fc2b, out);
}